// MyLSTM_20753281974596
// MI455X (gfx1250) — compile-verified
//
#include <hip/hip_runtime.h>
#include <stdint.h>

// ---------------------------------------------------------------------------
// Bidirectional LSTM, L=2048, D=H=1024, f32.
// Phase A: gx[dir][t][4H] = x_row(dir,t) . Wih_dir^T + b_dir  via f32 WMMA.
// Phase B: persistent recurrence kernel, Whh resident in VGPRs, grid barrier
//          per timestep per direction.
// ---------------------------------------------------------------------------

#define L_SEQ 2048
#define DDIM  1024
#define HDIM  1024
#define G4    4096          // 4*H

typedef __attribute__((ext_vector_type(2))) float v2f;
typedef __attribute__((ext_vector_type(8))) float v8f;

// ===========================================================================
// Kernel A: WMMA f32 GEMM.  Row space = 2 dirs x 2048 steps (dir1 uses
// reversed x rows).  Each wave computes a 16(M=steps) x 64(N=gate rows) tile.
// A frag: 16x4 f32 (lane m=lane&15, k-half=lane>>4, 2 f32/lane).
// B frag: 4x16 f32 (lane n=lane&15, k-half=lane>>4, 2 f32/lane).
// C/D   : 16x16 f32, VGPR v: M = v + 8*(lane>>4), N = lane&15.
// ===========================================================================
__global__ __launch_bounds__(256)
void lstm_gx_wmma(const float* __restrict__ x,
                  const float* __restrict__ wih,
                  const float* __restrict__ bias,
                  const float* __restrict__ wih_r,
                  const float* __restrict__ bias_r,
                  float* __restrict__ gx)
{
    const int wave = blockIdx.x * 8 + (threadIdx.x >> 5);
    const int lane = threadIdx.x & 31;

    const int rt = wave >> 6;          // 0..255 row tiles (16 steps each)
    const int cg = wave & 63;          // 0..63  col groups (64 gate rows each)
    const int d  = rt >> 7;            // direction
    const int t0 = (rt & 127) * 16;    // first recurrence step of tile
    const int r0 = cg * 64;            // first gate row of tile

    const int n   = lane & 15;         // M for A-frag, N for B/C-frags
    const int kh  = lane >> 4;         // which K-half this lane holds
    const int kh2 = kh * 2;

    const float* wih_d  = d ? wih_r  : wih;
    const float* bias_d = d ? bias_r : bias;

    // A source row: dir1 runs the sequence reversed.
    const int step = t0 + n;
    const int xr   = d ? (L_SEQ - 1 - step) : step;
    const float* aptr = x + (size_t)xr * DDIM + kh2;

    const float* bptr[4];
    v8f acc[4];
#pragma unroll
    for (int j = 0; j < 4; ++j) {
        bptr[j] = wih_d + (size_t)(r0 + j * 16 + n) * DDIM + kh2;
        const float bz = bias_d[r0 + j * 16 + n];
        acc[j] = (v8f){bz, bz, bz, bz, bz, bz, bz, bz};
    }

    for (int k = 0; k < DDIM; k += 4) {
        const v2f a = *(const v2f*)(aptr + k);
#pragma unroll
        for (int j = 0; j < 4; ++j) {
            const v2f b = *(const v2f*)(bptr[j] + k);
            acc[j] = __builtin_amdgcn_wmma_f32_16x16x4_f32(
                false, a, false, b, (short)0, acc[j], false, false);
        }
    }

    // Store: gx laid out [dir][t][4H].
#pragma unroll
    for (int j = 0; j < 4; ++j) {
        union { v8f v; float f[8]; } u; u.v = acc[j];
#pragma unroll
        for (int v = 0; v < 8; ++v) {
            const int m = v + 8 * kh;                      // step within tile
            gx[((size_t)(d * L_SEQ + t0 + m)) * G4 + r0 + j * 16 + n] = u.f[v];
        }
    }
}

// ===========================================================================
// Kernel B: persistent recurrence.
//   256 blocks = 2 dirs x 128 blocks.  Block owns 8 h-channels => 32 gate
//   rows.  256 threads: tid = row*8 + part; each thread keeps 128 Whh f32
//   in VGPRs (fully unrolled array) for the whole sequence.
// ===========================================================================
#define NBLK 128           // blocks per direction
#define CH   8             // h-channels per block

__device__ __forceinline__ float sigf(float v) {
    return 1.0f / (1.0f + __expf(-v));
}
__device__ __forceinline__ float tanh_fast(float v) {
    const float e = __expf(-2.0f * fabsf(v));
    const float t = (1.0f - e) / (1.0f + e);
    return copysignf(t, v);
}

__device__ __forceinline__ void grid_barrier(unsigned* cnt, unsigned* gen, int nb)
{
    __syncthreads();
    if (threadIdx.x == 0) {
        __threadfence();
        const unsigned g = __hip_atomic_load(gen, __ATOMIC_ACQUIRE,
                                             __HIP_MEMORY_SCOPE_AGENT);
        const unsigned old = __hip_atomic_fetch_add(cnt, 1u, __ATOMIC_ACQ_REL,
                                                    __HIP_MEMORY_SCOPE_AGENT);
        if (old == (unsigned)(nb - 1)) {
            __hip_atomic_store(cnt, 0u, __ATOMIC_RELAXED,
                               __HIP_MEMORY_SCOPE_AGENT);
            __hip_atomic_fetch_add(gen, 1u, __ATOMIC_RELEASE,
                                   __HIP_MEMORY_SCOPE_AGENT);
        } else {
            while (__hip_atomic_load(gen, __ATOMIC_ACQUIRE,
                                     __HIP_MEMORY_SCOPE_AGENT) == g)
                __builtin_amdgcn_s_sleep(1);
        }
        __threadfence();
    }
    __syncthreads();
}

__global__ __launch_bounds__(256)
void lstm_recurrence(const float* __restrict__ gx,
                     const float* __restrict__ whh,
                     const float* __restrict__ whh_r,
                     float* __restrict__ hb,       // [2 dirs][2 bufs][H]
                     unsigned* __restrict__ ctrl,  // cnt[2], gen[2]
                     float* __restrict__ out)      // [L][2H]
{
    __shared__ float lh[8 * 132];   // h staged with 4-float pad per 128 chunk
    __shared__ float gacc[32];      // one partial-reduced gate value per row

    const int tid  = threadIdx.x;
    const int d    = blockIdx.x >> 7;       // direction
    const int bh   = blockIdx.x & (NBLK - 1);
    const int k0   = bh * CH;               // first owned h-channel
    const int row  = tid >> 3;              // 0..31 gate row within block
    const int part = tid & 7;               // K partition (128 each)
    const int g    = row >> 3;              // gate 0..3
    const int cc   = row & 7;               // channel within block

    // --- load this thread's 128 Whh weights into registers (persistent) ---
    const float* whh_d = d ? whh_r : whh;
    const float* wsrc  = whh_d + ((size_t)(g * HDIM + k0 + cc)) * HDIM + part * 128;
    float w[128];
#pragma unroll
    for (int i = 0; i < 128; i += 4) {
        const float4 v = *(const float4*)(wsrc + i);
        w[i] = v.x; w[i + 1] = v.y; w[i + 2] = v.z; w[i + 3] = v.w;
    }

    // --- init h(t=0)=0 for owned channels; c kept per-thread (tid<CH) ---
    float c_reg = 0.0f;
    if (tid < CH) hb[(d * 2 + 0) * HDIM + k0 + tid] = 0.0f;
    grid_barrier(&ctrl[d], &ctrl[2 + d], NBLK);

    for (int t = 0; t < L_SEQ; ++t) {
        // stage h(t) into padded LDS (1024 floats, 4 per thread)
        {
            const float* hsrc = hb + (d * 2 + (t & 1)) * HDIM;
            const float4 hv = *(const float4*)(hsrc + tid * 4);
            const int pos = (tid >> 5) * 132 + ((tid * 4) & 127);
            lh[pos] = hv.x; lh[pos + 1] = hv.y;
            lh[pos + 2] = hv.z; lh[pos + 3] = hv.w;
        }
        __syncthreads();

        // 128 register-resident FMAs against the staged h slice
        float sum = 0.0f;
        const int base = part * 132;
#pragma unroll
        for (int i = 0; i < 128; i += 4) {
            const float4 hv = *(const float4*)(&lh[base + i]);
            sum = fmaf(w[i],     hv.x, sum);
            sum = fmaf(w[i + 1], hv.y, sum);
            sum = fmaf(w[i + 2], hv.z, sum);
            sum = fmaf(w[i + 3], hv.w, sum);
        }
        // reduce the 8 K-partitions (consecutive lanes)
        sum += __shfl_xor(sum, 1, 32);
        sum += __shfl_xor(sum, 2, 32);
        sum += __shfl_xor(sum, 4, 32);
        if (part == 0) gacc[row] = sum;
        __syncthreads();

        if (tid < CH) {
            const float* gxt = gx + ((size_t)(d * L_SEQ + t)) * G4 + k0 + tid;
            const float gi = gacc[tid]          + gxt[0];
            const float gf = gacc[CH + tid]     + gxt[HDIM];
            const float gg = gacc[2 * CH + tid] + gxt[2 * HDIM];
            const float go = gacc[3 * CH + tid] + gxt[3 * HDIM];
            c_reg = sigf(gf) * c_reg + sigf(gi) * tanh_fast(gg);
            const float hval = sigf(go) * tanh_fast(c_reg);
            hb[(d * 2 + ((t + 1) & 1)) * HDIM + k0 + tid] = hval;
            const int tt = d ? (L_SEQ - 1 - t) : t;
            out[(size_t)tt * (2 * HDIM) + d * HDIM + k0 + tid] = hval;
        }
        grid_barrier(&ctrl[d], &ctrl[2 + d], NBLK);
    }
}

// ===========================================================================
// Launcher.  Workspace layout:
//   [0,   64)        barrier control (zeroed by async memset each call)
//   [64,  16448)     h double buffers: 2 dirs x 2 bufs x 1024 f32
//   [64K, 64K+64MB)  gx: 2 x 2048 x 4096 f32
// ===========================================================================
extern "C" void kernel_launch(void* const* d_in, const int* in_sizes, int n_in,
                              void* d_out, int out_size, void* d_ws, size_t ws_size,
                              hipStream_t stream)
{
    (void)in_sizes; (void)n_in; (void)out_size; (void)ws_size;

    const float* x      = (const float*)d_in[0];
    const float* wih    = (const float*)d_in[1];
    const float* whh    = (const float*)d_in[2];
    const float* bias   = (const float*)d_in[3];
    const float* wih_r  = (const float*)d_in[4];
    const float* whh_r  = (const float*)d_in[5];
    const float* bias_r = (const float*)d_in[6];
    float* out = (float*)d_out;

    uint8_t*  ws   = (uint8_t*)d_ws;
    unsigned* ctrl = (unsigned*)ws;
    float*    hb   = (float*)(ws + 64);
    float*    gx   = (float*)(ws + 65536);

    hipMemsetAsync(d_ws, 0, 64, stream);

    // Phase A: 256 row tiles x 64 col groups = 16384 waves = 2048 blocks.
    lstm_gx_wmma<<<2048, 256, 0, stream>>>(x, wih, bias, wih_r, bias_r, gx);

    // Phase B: 2 dirs x 128 persistent blocks.
    lstm_recurrence<<<2 * NBLK, 256, 0, stream>>>(gx, whh, whh_r, hb, ctrl, out);
}